// LaunderingGNN_41171556499595
// MI455X (gfx1250) — compile-verified
//
#include <hip/hip_runtime.h>
#include <hip/hip_bf16.h>

typedef __attribute__((ext_vector_type(2))) float v2f;
typedef __attribute__((ext_vector_type(8))) float v8f;

#define IN_FEAT 64

// ---------------------------------------------------------------------------
// degree count: one thread per edge
// ---------------------------------------------------------------------------
__global__ __launch_bounds__(256)
void sage_deg_kernel(const int* __restrict__ dst, float* __restrict__ deg, int E) {
    int e = blockIdx.x * blockDim.x + threadIdx.x;
    if (e < E) atomicAdd(&deg[dst[e]], 1.0f);
}

// ---------------------------------------------------------------------------
// scatter-add of 64-float feature rows: one wave32 per edge, 2 floats/lane
// ---------------------------------------------------------------------------
__global__ __launch_bounds__(256)
void sage_scatter_kernel(const float* __restrict__ feat,
                         const int* __restrict__ src,
                         const int* __restrict__ dst,
                         float* __restrict__ agg, int E) {
    int t    = blockIdx.x * blockDim.x + threadIdx.x;
    int e    = t >> 5;
    int lane = t & 31;
    if (e >= E) return;
    int s = src[e];
    int d = dst[e];
    const float2 v = *(const float2*)(feat + (size_t)s * IN_FEAT + lane * 2);
    float* a = agg + (size_t)d * IN_FEAT + lane * 2;
    atomicAdd(&a[0], v.x);
    atomicAdd(&a[1], v.y);
}

// ---------------------------------------------------------------------------
// Fused SAGE update: out = act( (agg/deg) @ Wl  +  xin @ Wr  +  b )
// One wave computes one 16x16 output tile with V_WMMA_F32_16X16X4_F32.
// K = 64 (IN_FEAT), OUT_DIM in {64, 32}. Weights staged in LDS.
// ---------------------------------------------------------------------------
template <int OUT_DIM, bool RELU>
__global__ __launch_bounds__(256)
void sage_gemm_kernel(const float* __restrict__ agg,
                      const float* __restrict__ deg,
                      const float* __restrict__ xin,
                      const float* __restrict__ Wl,
                      const float* __restrict__ Wr,
                      const float* __restrict__ bias,
                      float* __restrict__ out, int n_tiles_m) {
    __shared__ float sWl[IN_FEAT * OUT_DIM];
    __shared__ float sWr[IN_FEAT * OUT_DIM];
    for (int i = threadIdx.x; i < IN_FEAT * OUT_DIM; i += 256) {
        sWl[i] = Wl[i];
        sWr[i] = Wr[i];
    }
    __syncthreads();

    constexpr int TILES_N = OUT_DIM / 16;
    int wave = (int)((blockIdx.x * 256u + threadIdx.x) >> 5);
    int lane = threadIdx.x & 31;
    if (wave >= n_tiles_m * TILES_N) return;   // uniform per-wave: EXEC stays all-1s

    int tm   = wave / TILES_N;
    int tn   = wave % TILES_N;
    int half = lane >> 4;     // 0: K pair {0,1}, 1: K pair {2,3} of each k-step
    int r    = lane & 15;

    int row = tm * 16 + r;    // node row this lane feeds into A
    int col = tn * 16 + r;    // output column this lane feeds into B

    float rdeg = 1.0f / fmaxf(deg[row], 1.0f);
    const float* arow = agg + (size_t)row * IN_FEAT;
    const float* xrow = xin + (size_t)row * IN_FEAT;

    v8f acc = {0.f, 0.f, 0.f, 0.f, 0.f, 0.f, 0.f, 0.f};

#pragma unroll
    for (int k0 = 0; k0 < IN_FEAT; k0 += 4) {
        int kk = k0 + half * 2;
        // A tiles: 16x4 fp32, one row per lane -> contiguous float2 loads
        v2f a1, a2;
        a1.x = arow[kk]     * rdeg;
        a1.y = arow[kk + 1] * rdeg;
        a2.x = xrow[kk];
        a2.y = xrow[kk + 1];
        // B tiles: 4x16 fp32 from LDS-staged weights (row-major [K][OUT_DIM])
        v2f bl, br;
        bl.x = sWl[kk * OUT_DIM + col];
        bl.y = sWl[(kk + 1) * OUT_DIM + col];
        br.x = sWr[kk * OUT_DIM + col];
        br.y = sWr[(kk + 1) * OUT_DIM + col];
        acc = __builtin_amdgcn_wmma_f32_16x16x4_f32(false, a1, false, bl,
                                                    (short)0, acc, false, false);
        acc = __builtin_amdgcn_wmma_f32_16x16x4_f32(false, a2, false, br,
                                                    (short)0, acc, false, false);
    }

    float bv = bias[col];
#pragma unroll
    for (int i = 0; i < 8; ++i) {
        float v = acc[i] + bv;
        if (RELU) v = fmaxf(v, 0.0f);
        out[(size_t)(tm * 16 + half * 8 + i) * OUT_DIM + col] = v;
    }
}

// ---------------------------------------------------------------------------
// Final linear 32 -> 2: one thread per node
// ---------------------------------------------------------------------------
__global__ __launch_bounds__(256)
void sage_out_kernel(const float* __restrict__ h2,
                     const float* __restrict__ W,   // [32][2]
                     const float* __restrict__ b,   // [2]
                     float* __restrict__ out, int N) {
    int n = blockIdx.x * blockDim.x + threadIdx.x;
    if (n >= N) return;
    const float* hr = h2 + (size_t)n * 32;
    float o0 = b[0], o1 = b[1];
#pragma unroll
    for (int k = 0; k < 32; ++k) {
        float h = hr[k];
        o0 = fmaf(h, W[k * 2 + 0], o0);
        o1 = fmaf(h, W[k * 2 + 1], o1);
    }
    out[(size_t)n * 2 + 0] = o0;
    out[(size_t)n * 2 + 1] = o1;
}

// ---------------------------------------------------------------------------
extern "C" void kernel_launch(void* const* d_in, const int* in_sizes, int n_in,
                              void* d_out, int out_size, void* d_ws, size_t ws_size,
                              hipStream_t stream) {
    const float* x    = (const float*)d_in[0];
    const int*   ei   = (const int*)d_in[1];     // [2][E] int32
    const float* Wl1  = (const float*)d_in[2];
    const float* Wr1  = (const float*)d_in[3];
    const float* b1   = (const float*)d_in[4];
    const float* Wl2  = (const float*)d_in[5];
    const float* Wr2  = (const float*)d_in[6];
    const float* b2   = (const float*)d_in[7];
    const float* Wout = (const float*)d_in[8];
    const float* bout = (const float*)d_in[9];

    const int N = in_sizes[0] / IN_FEAT;   // 100000 (divisible by 16)
    const int E = in_sizes[1] / 2;         // 1000000
    const int* src = ei;
    const int* dst = ei + E;

    // workspace carve (floats): deg[N] | agg[N*64] | h[N*64] | h2[N*32]
    float* ws  = (float*)d_ws;
    float* deg = ws;                 ws += N;
    float* agg = ws;                 ws += (size_t)N * 64;
    float* h   = ws;                 ws += (size_t)N * 64;
    float* h2  = ws;
    float* out = (float*)d_out;

    const int tilesM = N / 16;

    // ---- layer 1 -----------------------------------------------------------
    hipMemsetAsync(deg, 0, (size_t)N * sizeof(float), stream);
    hipMemsetAsync(agg, 0, (size_t)N * 64 * sizeof(float), stream);
    sage_deg_kernel<<<(E + 255) / 256, 256, 0, stream>>>(dst, deg, E);
    sage_scatter_kernel<<<(E * 32 + 255) / 256, 256, 0, stream>>>(x, src, dst, agg, E);
    {
        int waves  = tilesM * (64 / 16);
        int blocks = (waves * 32 + 255) / 256;
        sage_gemm_kernel<64, true><<<blocks, 256, 0, stream>>>(
            agg, deg, x, Wl1, Wr1, b1, h, tilesM);
    }

    // ---- layer 2 -----------------------------------------------------------
    hipMemsetAsync(agg, 0, (size_t)N * 64 * sizeof(float), stream);
    sage_scatter_kernel<<<(E * 32 + 255) / 256, 256, 0, stream>>>(h, src, dst, agg, E);
    {
        int waves  = tilesM * (32 / 16);
        int blocks = (waves * 32 + 255) / 256;
        sage_gemm_kernel<32, false><<<blocks, 256, 0, stream>>>(
            agg, deg, h, Wl2, Wr2, b2, h2, tilesM);
    }

    // ---- output linear -----------------------------------------------------
    sage_out_kernel<<<(N + 255) / 256, 256, 0, stream>>>(h2, Wout, bout, out, N);
}